// LI_88098369175836
// MI455X (gfx1250) — compile-verified
//
#include <hip/hip_runtime.h>

// Leaky-integrator scan u_t = TAU*u_{t-1} + x_t over the last (contiguous) axis.
// Formulated as tiled triangular GEMM on the CDNA5 fp32 matrix pipe:
//   U(16x16) = X(16x16) * T(16x16) + Carry,  T[s][t] = TAU^(t-s) for s<=t
// done as 4 chained V_WMMA_F32_16X16X4_F32 (K=4 each), carry folded into C.

typedef __attribute__((ext_vector_type(2))) float v2f;
typedef __attribute__((ext_vector_type(8))) float v8f;

#define TAU 0.9f
#define T_STEPS 2048
#define CHUNK 16
#define N_CHUNKS (T_STEPS / CHUNK)
#define ROWS_PER_WAVE 16
#define WAVES_PER_BLOCK 8

// tau^e for e in [0,15], exact products via bit-select (constant-folded t1..t8).
__device__ __forceinline__ float tau_pow_u4(int e) {
  const float t1 = TAU;
  const float t2 = t1 * t1;
  const float t4 = t2 * t2;
  const float t8 = t4 * t4;
  float p = (e & 1) ? t1 : 1.0f;
  p *= (e & 2) ? t2 : 1.0f;
  p *= (e & 4) ? t4 : 1.0f;
  p *= (e & 8) ? t8 : 1.0f;
  return p;
}

__global__ __launch_bounds__(256) void li_scan_wmma_f32(
    const float* __restrict__ x, float* __restrict__ out, int n_row_groups) {
  const int lane  = threadIdx.x & 31;
  const int wave  = threadIdx.x >> 5;
  const int group = blockIdx.x * WAVES_PER_BLOCK + wave;
  if (group >= n_row_groups) return;  // wave-uniform; grid exactly covers groups

  const int hi = lane >> 4;   // half-wave select
  const int n  = lane & 15;   // N (time col) for B/C/D; M (row) for A loads

  // B fragments: 4 K-blocks of the 16x16 triangular tau matrix.
  // B(4x16) layout: VGPR j -> row K=j (lanes 0-15), K=j+2 (lanes 16-31), N = lane&15.
  v2f bfrag[4];
#pragma unroll
  for (int blk = 0; blk < 4; ++blk) {
#pragma unroll
    for (int j = 0; j < 2; ++j) {
      const int s = 4 * blk + j + 2 * hi;          // source time within chunk
      bfrag[blk][j] = (s <= n) ? tau_pow_u4(n - s) : 0.0f;
    }
  }

  const float carry_coef = TAU * tau_pow_u4(n);    // tau^(n+1)
  const int   carry_src  = hi ? 31 : 15;           // lane holding N=15 of prev D

  const long long row0 = (long long)group * ROWS_PER_WAVE;
  // A-load base: lane loads X[row0 + n][t0 + 4*blk + 2*hi .. +1] as float2.
  const float* __restrict__ xlane = x + (row0 + (long long)n) * T_STEPS + 2 * hi;

  v8f d = {};  // D of previous chunk == u state at tile end; u_0 = 0

  // Register double-buffer for A tiles so loads overlap the WMMA chain.
  v2f a_cur[4], a_nxt[4];
#pragma unroll
  for (int blk = 0; blk < 4; ++blk)
    a_cur[blk] = *(const v2f*)(xlane + 4 * blk);

  for (int t = 0; t < N_CHUNKS; ++t) {
    const int t0 = t * CHUNK;
    if (t + 1 < N_CHUNKS) {
#pragma unroll
      for (int blk = 0; blk < 4; ++blk)
        a_nxt[blk] = *(const v2f*)(xlane + t0 + CHUNK + 4 * blk);
    }

    // Carry term into accumulator: c[r] = tau^(n+1) * u_prev[M],
    // u_prev[M] = previous D at N=15 (lane 15 for M=r, lane 31 for M=r+8).
    v8f c;
#pragma unroll
    for (int r = 0; r < 8; ++r)
      c[r] = carry_coef * __shfl(d[r], carry_src, 32);
    d = c;

    // U = X * T + C via 4 chained fp32 WMMAs (K=4 each).
#pragma unroll
    for (int blk = 0; blk < 4; ++blk)
      d = __builtin_amdgcn_wmma_f32_16x16x4_f32(
          /*neg_a=*/false, a_cur[blk],
          /*neg_b=*/false, bfrag[blk],
          /*c_mod=*/(short)0, d,
          /*reuse_a=*/false, /*reuse_b=*/false);

    // Store D: VGPR r -> out[(row0 + r + 8*hi) * T + t0 + n]
#pragma unroll
    for (int r = 0; r < 8; ++r)
      out[(row0 + r + 8 * hi) * (long long)T_STEPS + t0 + n] = d[r];

#pragma unroll
    for (int blk = 0; blk < 4; ++blk)
      a_cur[blk] = a_nxt[blk];
  }
}

extern "C" void kernel_launch(void* const* d_in, const int* in_sizes, int n_in,
                              void* d_out, int out_size, void* d_ws, size_t ws_size,
                              hipStream_t stream) {
  (void)n_in; (void)d_ws; (void)ws_size; (void)out_size;
  const float* x = (const float*)d_in[0];
  float* out = (float*)d_out;

  const long long total = (long long)in_sizes[0];        // 32*1024*2048
  const int rows = (int)(total / T_STEPS);               // 32768
  const int groups = rows / ROWS_PER_WAVE;               // 2048 waves
  const int blocks = (groups + WAVES_PER_BLOCK - 1) / WAVES_PER_BLOCK;  // 256

  li_scan_wmma_f32<<<blocks, 32 * WAVES_PER_BLOCK, 0, stream>>>(x, out, groups);
}